// DigitCapsule_29918742184695
// MI455X (gfx1250) — compile-verified
//
#include <hip/hip_runtime.h>
#include <math.h>

typedef __attribute__((ext_vector_type(2))) float v2f;
typedef __attribute__((ext_vector_type(4))) float v4f;
typedef __attribute__((ext_vector_type(8))) float v8f;

#define N1    1152
#define D1    8
#define C2    10
#define D2    16
#define BATCH 512
#define NCHUNK 8
#define NPER  (N1 / NCHUNK)      /* 144 */
#define BT    (BATCH / 16)       /* 32 batch tiles */

// D = A(16x4) * B(4x16) + C, fp32. M = d (digit-caps dim), N = batch-in-tile.
// K-permutation trick: lane(li,half) loads ONE contiguous v4f = {k0..k0+3},
// k0 = 4*half. frag0 = elements {0,1} (global k {0,1}|{4,5}),
// frag1 = elements {2,3} (global k {2,3}|{6,7}). A and B use the same
// slot->k mapping, so the K contraction is exact; each k counted once.
// C/D lane(li,half) vgpr j: element (d = j + 8*half, m = li)

__device__ __forceinline__ v8f wmma4(v2f a, v2f b, v8f c) {
  return __builtin_amdgcn_wmma_f32_16x16x4_f32(false, a, false, b, (short)0, c,
                                               false, false);
}

__device__ __forceinline__ v2f lo2(v4f v) { return __builtin_shufflevector(v, v, 0, 1); }
__device__ __forceinline__ v2f hi2(v4f v) { return __builtin_shufflevector(v, v, 2, 3); }

__device__ __forceinline__ float squash_scale(float sq) {
  // |v|^2/(1+|v|^2) / |v|
  return (sq / (1.0f + sq)) * rsqrtf(sq);
}

// ---------------- K1: s1[b,c,d] = sum_{n,k} w[c,n,d,k] * x[b,n,k] ----------------
__global__ __launch_bounds__(128) void k1_s1(const float* __restrict__ x,
                                             const float* __restrict__ w,
                                             float* __restrict__ s1) {
  const int wave = blockIdx.x * 4 + (threadIdx.x >> 5);   // 2560 waves
  const int lane = threadIdx.x & 31;
  const int li = lane & 15, half = lane >> 4;
  const int chunk = wave & (NCHUNK - 1);
  const int c     = (wave >> 3) % C2;
  const int bt    = wave / (C2 * NCHUNK);
  const int b0    = bt * 16;

  const float* wp = w + ((size_t)c * N1 * D2 + li) * D1 + 4 * half;   // + n*128
  const float* xp = x + (size_t)(b0 + li) * N1 * D1 + 4 * half;       // + n*8

  v8f acc = {};
  const int n0 = chunk * NPER;
  for (int n = n0; n < n0 + NPER; ++n) {
    v4f aw = *(const v4f*)(wp + (size_t)n * (D2 * D1));   // one b128 per lane
    v4f bx = *(const v4f*)(xp + (size_t)n * D1);          // one b128 per lane
    acc = wmma4(lo2(aw), lo2(bx), acc);
    acc = wmma4(hi2(aw), hi2(bx), acc);
  }
  float* sp = s1 + ((size_t)(b0 + li) * C2 + c) * D2 + 8 * half;
#pragma unroll
  for (int j = 0; j < 8; ++j) unsafeAtomicAdd(sp + j, acc[j]);
}

// ---------------- K2: o1 = squash(0.1 * s1)  (iter-1 outputs, c uniform) --------
__global__ void k2_squash(const float* __restrict__ s1, float* __restrict__ o1) {
  int t = blockIdx.x * blockDim.x + threadIdx.x;          // (b*10 + c)
  if (t >= BATCH * C2) return;
  const float* sp = s1 + (size_t)t * D2;
  float v[D2], sq = 0.0f;
#pragma unroll
  for (int d = 0; d < D2; ++d) { v[d] = sp[d] * 0.1f; sq += v[d] * v[d]; }
  float sc = squash_scale(sq);
  float* op = o1 + (size_t)t * D2;
#pragma unroll
  for (int d = 0; d < D2; ++d) op[d] = v[d] * sc;
}

// ---------------- K3: fused routing iter 2 --------------------------------------
// per (b,n): h[c,d] = x_hat (WMMA), logit p[c] = <o1[b,c,:], h[c,:]>,
// c2 = softmax_c(p), acc[b,c,d] += c2[c] * h[c,d]
__global__ __launch_bounds__(128) void k3_route(const float* __restrict__ x,
                                                const float* __restrict__ w,
                                                const float* __restrict__ o1,
                                                float* __restrict__ accg) {
  const int wave = blockIdx.x * 4 + (threadIdx.x >> 5);   // 256 waves
  const int lane = threadIdx.x & 31;
  const int li = lane & 15, half = lane >> 4;
  const int chunk = wave & (NCHUNK - 1);
  const int bt    = wave >> 3;
  const int b0    = bt * 16;

  // lane-resident slice of o1: o1[b0+li, c, (8*half)+j], j=0..7  (32B aligned)
  v8f o1r[C2];
#pragma unroll
  for (int c = 0; c < C2; ++c)
    o1r[c] = *(const v8f*)(o1 + ((size_t)(b0 + li) * C2 + c) * D2 + 8 * half);

  v8f acc[C2];
#pragma unroll
  for (int c = 0; c < C2; ++c) acc[c] = (v8f){};

  const float* xp = x + (size_t)(b0 + li) * N1 * D1 + 4 * half;
  const int n0 = chunk * NPER;
  for (int n = n0; n < n0 + NPER; ++n) {
    v4f bx = *(const v4f*)(xp + (size_t)n * D1);          // one b128 per lane
    v2f bx0 = lo2(bx), bx1 = hi2(bx);

    v8f h[C2];
    float p[C2];
#pragma unroll
    for (int c = 0; c < C2; ++c) {
      const float* wn = w + (((size_t)c * N1 + n) * D2 + li) * D1 + 4 * half;
      v4f aw = *(const v4f*)(wn);                         // one b128 per lane
      v8f hz = {};
      hz = wmma4(lo2(aw), bx0, hz);
      hz = wmma4(hi2(aw), bx1, hz);
      h[c] = hz;
      float pc = 0.0f;
#pragma unroll
      for (int j = 0; j < 8; ++j) pc += hz[j] * o1r[c][j];
      pc += __shfl_xor(pc, 16, 32);   // merge the two d-halves -> full logit
      p[c] = pc;
    }
    // softmax over the 10 capsules (lane-local)
    float m = p[0];
#pragma unroll
    for (int c = 1; c < C2; ++c) m = fmaxf(m, p[c]);
    float s = 0.0f, e[C2];
#pragma unroll
    for (int c = 0; c < C2; ++c) { e[c] = __expf(p[c] - m); s += e[c]; }
    float inv = 1.0f / s;
#pragma unroll
    for (int c = 0; c < C2; ++c) {
      float c2v = e[c] * inv;
#pragma unroll
      for (int j = 0; j < 8; ++j) acc[c][j] += c2v * h[c][j];
    }
  }

#pragma unroll
  for (int c = 0; c < C2; ++c) {
    float* ap = accg + ((size_t)(b0 + li) * C2 + c) * D2 + 8 * half;
#pragma unroll
    for (int j = 0; j < 8; ++j) unsafeAtomicAdd(ap + j, acc[c][j]);
  }
}

// ---------------- K4: out = squash(acc) -----------------------------------------
__global__ void k4_squash(const float* __restrict__ accg, float* __restrict__ out) {
  int t = blockIdx.x * blockDim.x + threadIdx.x;          // (b*10 + c)
  if (t >= BATCH * C2) return;
  const float* ap = accg + (size_t)t * D2;
  float v[D2], sq = 0.0f;
#pragma unroll
  for (int d = 0; d < D2; ++d) { v[d] = ap[d]; sq += v[d] * v[d]; }
  float sc = squash_scale(sq);
  float* op = out + (size_t)t * D2;
#pragma unroll
  for (int d = 0; d < D2; ++d) op[d] = v[d] * sc;
}

extern "C" void kernel_launch(void* const* d_in, const int* in_sizes, int n_in,
                              void* d_out, int out_size, void* d_ws, size_t ws_size,
                              hipStream_t stream) {
  (void)in_sizes; (void)n_in; (void)out_size; (void)ws_size;
  const float* x = (const float*)d_in[0];     // [512,1152,8]
  const float* w = (const float*)d_in[1];     // [10,1152,16,8]
  float* out = (float*)d_out;                 // [512,10,16]

  const size_t NE = (size_t)BATCH * C2 * D2;  // 81920
  float* s1  = (float*)d_ws;                  // [512,10,16]
  float* acc = s1 + NE;                       // [512,10,16]
  float* o1  = acc + NE;                      // [512,10,16]

  // s1 and acc are accumulated with atomics -> zero them (graph-capturable).
  hipMemsetAsync(d_ws, 0, 2 * NE * sizeof(float), stream);

  // K1: 32 btiles * 10 c * 8 nchunks = 2560 waves, 4 waves/block
  k1_s1<<<dim3(BT * C2 * NCHUNK / 4), dim3(128), 0, stream>>>(x, w, s1);
  // K2: 5120 (b,c) pairs
  k2_squash<<<dim3((BATCH * C2 + 127) / 128), dim3(128), 0, stream>>>(s1, o1);
  // K3: 32 btiles * 8 nchunks = 256 waves, 4 waves/block
  k3_route<<<dim3(BT * NCHUNK / 4), dim3(128), 0, stream>>>(x, w, o1, acc);
  // K4: final squash to output
  k4_squash<<<dim3((BATCH * C2 + 127) / 128), dim3(128), 0, stream>>>(acc, out);
}